// SegFormerHead_87711822119598
// MI455X (gfx1250) — compile-verified
//
#include <hip/hip_runtime.h>
#include <hip/hip_bf16.h>

typedef float v2f __attribute__((ext_vector_type(2)));
typedef float v8f __attribute__((ext_vector_type(8)));
typedef int   v4i __attribute__((ext_vector_type(4)));

#if __has_builtin(__builtin_amdgcn_global_load_async_to_lds_b128) && \
    __has_builtin(__builtin_amdgcn_s_wait_asynccnt)
#define USE_ASYNC_LDS 1
#else
#define USE_ASYNC_LDS 0
#endif

// ---------------------------------------------------------------------------
// Generic fp32 WMMA GEMM:  Out[b][p][n] = sum_c A[b][c][p] (or A[b][p][c]) * W[c][n] + bias[n]
// A 16x4 fragment layout (ISA 7.12.2): lane<16 -> M=lane, K={c0,c0+1}; lane>=16 -> M=lane-16, K={c0+2,c0+3}
// Block = 256 threads = 8 waves; wave handles one 16-pixel tile x (16*NTW) outputs.
// ---------------------------------------------------------------------------
template <bool A_CH_MAJOR, int NTW>
__global__ void wmma_gemm_kernel(const float* __restrict__ X, long xBatchStride,
                                 int P, int ldA,
                                 const float* __restrict__ Wm,
                                 const float* __restrict__ bias,
                                 float* __restrict__ Out, long outBatchStride, int outStride,
                                 int K, int N) {
    const int tid  = threadIdx.x;
    const int wave = tid >> 5;
    const int lane = tid & 31;
    const int lo   = lane & 15;
    const int hi   = lane >> 4;

    const int b  = blockIdx.z;
    const int p0 = (blockIdx.x * 8 + wave) * 16;
    const int n_base = blockIdx.y * (16 * NTW);

    const float* Xb = X + (long)b * xBatchStride;

    v8f acc[NTW] = {};

    for (int c0 = 0; c0 < K; c0 += 4) {
        v2f a;
        if (A_CH_MAJOR) {
            a.x = Xb[(long)(c0 + 2 * hi) * P + p0 + lo];
            a.y = Xb[(long)(c0 + 2 * hi + 1) * P + p0 + lo];
        } else {
            const float* ap = Xb + (long)(p0 + lo) * ldA + c0 + 2 * hi;
            a.x = ap[0];
            a.y = ap[1];
        }
#pragma unroll
        for (int t = 0; t < NTW; ++t) {
            const int n = n_base + t * 16 + lo;
            v2f bb;
            bb.x = Wm[(long)(c0 + 2 * hi) * N + n];
            bb.y = Wm[(long)(c0 + 2 * hi + 1) * N + n];
            acc[t] = __builtin_amdgcn_wmma_f32_16x16x4_f32(
                false, a, false, bb, (short)0, acc[t], false, false);
        }
    }

    float* Ob = Out + (long)b * outBatchStride;
#pragma unroll
    for (int t = 0; t < NTW; ++t) {
        const int n  = n_base + t * 16 + lo;
        const float bv = bias ? bias[n] : 0.0f;
#pragma unroll
        for (int r = 0; r < 8; ++r) {
            const int p = p0 + r + 8 * hi;
            Ob[(long)p * outStride + n] = acc[t][r] + bv;
        }
    }
}

// ---------------------------------------------------------------------------
// DySample: bilinear gather with border clamp.
// proj: [b][Hin*Win][384] (pixel-major), offs: [b][Hin*Win][8*s*s]
// grid_sample algebra reduces to sample = clamp(w + off, 0, W-1).
// ---------------------------------------------------------------------------
__global__ void dysample_kernel(const float* __restrict__ proj,
                                const float* __restrict__ offs,
                                float* __restrict__ cc,
                                int Hin, int Win, int s, int choff) {
    const long tid = (long)blockIdx.x * blockDim.x + threadIdx.x;
    const int c  = (int)(tid % 96);
    long r1 = tid / 96;
    const int gg = (int)(r1 & 3);
    long r2 = r1 >> 2;
    const int X = (int)(r2 & 127);
    const int Y = (int)((r2 >> 7) & 127);
    const int b = (int)(r2 >> 14);

    const int h = Y / s, i = Y % s;
    const int w = X / s, j = X % s;
    const int Nc = 8 * s * s;

    const long obase = ((long)b * Hin * Win + (long)h * Win + w) * Nc;
    const float hh_j = (j - (s - 1) * 0.5f) / (float)s;
    const float hh_i = (i - (s - 1) * 0.5f) / (float)s;
    const int ox = (gg * s + i) * s + j;
    const int oy = ((4 + gg) * s + i) * s + j;
    const float offx = offs[obase + ox] * 0.25f + hh_j;
    const float offy = offs[obase + oy] * 0.25f + hh_i;

    float xf = fminf(fmaxf((float)w + offx, 0.0f), (float)(Win - 1));
    float yf = fminf(fmaxf((float)h + offy, 0.0f), (float)(Hin - 1));
    const float x0f = floorf(xf), y0f = floorf(yf);
    const float wx = xf - x0f, wy = yf - y0f;
    const int x0 = (int)x0f, y0 = (int)y0f;
    const int x1 = min(x0 + 1, Win - 1), y1 = min(y0 + 1, Hin - 1);

    const long pb = (long)b * Hin * Win;
    const int cidx = gg * 96 + c;
    const float v00 = proj[(pb + (long)y0 * Win + x0) * 384 + cidx];
    const float v01 = proj[(pb + (long)y0 * Win + x1) * 384 + cidx];
    const float v10 = proj[(pb + (long)y1 * Win + x0) * 384 + cidx];
    const float v11 = proj[(pb + (long)y1 * Win + x1) * 384 + cidx];

    const float val = v00 * (1.0f - wx) * (1.0f - wy) + v01 * wx * (1.0f - wy) +
                      v10 * (1.0f - wx) * wy + v11 * wx * wy;

    cc[((long)b * 16384 + (long)Y * 128 + X) * 1536 + choff + cidx] = val;
}

// ---------------------------------------------------------------------------
// Channel mean over pixels (two stage).
// ---------------------------------------------------------------------------
__global__ void colsum_partial_kernel(const float* __restrict__ cc, float* __restrict__ partial) {
    const int c = blockIdx.x * 256 + threadIdx.x;   // 0..1535
    const int b = blockIdx.z;
    const long base = ((long)b * 16384 + (long)blockIdx.y * 1024) * 1536 + c;
    float sum = 0.0f;
    for (int i = 0; i < 1024; ++i) sum += cc[base + (long)i * 1536];
    partial[((long)blockIdx.y * 2 + b) * 1536 + c] = sum;
}

__global__ void colsum_reduce_kernel(const float* __restrict__ partial, float* __restrict__ s) {
    const int idx = blockIdx.x * 256 + threadIdx.x;
    if (idx >= 3072) return;
    const int b = idx / 1536, c = idx % 1536;
    float sum = 0.0f;
    for (int k = 0; k < 16; ++k) sum += partial[((long)k * 2 + b) * 1536 + c];
    s[idx] = sum * (1.0f / 16384.0f);
}

// ---------------------------------------------------------------------------
// SSM scan.  Rank-1 trick: u[c][d] = s[c]*in_w[d]+in_b[d]  =>
// proj[c][j] = s[c]*P1[j] + P0[j] with P1 = in_w @ xproj, P0 = in_b @ xproj.
// 1024 threads = (d 0..63, n 0..15); grid.x = 4 (b in {0,1} x dir in {0,1}).
// ---------------------------------------------------------------------------
__global__ void ssm_scan_kernel(const float* __restrict__ s,
                                const float* __restrict__ in_w, const float* __restrict__ in_b,
                                const float* __restrict__ xproj_w,   // [64][33]
                                const float* __restrict__ dt_w, const float* __restrict__ dt_b,
                                const float* __restrict__ A_log,     // [64][16]
                                float* __restrict__ ybuf)            // [4][1536][64]
{
    const int b = blockIdx.x & 1;
    const int dir = blockIdx.x >> 1;
    const int t = threadIdx.x;
    const int d = t >> 4;
    const int n = t & 15;

    __shared__ float P1s[33], P0s[33], s_sh[1536];
    if (t < 66) {
        const int j = (t < 33) ? t : t - 33;
        const bool isP1 = t < 33;
        float acc = 0.0f;
        for (int dd = 0; dd < 64; ++dd)
            acc = fmaf(isP1 ? in_w[dd] : in_b[dd], xproj_w[dd * 33 + j], acc);
        if (isP1) P1s[j] = acc; else P0s[j] = acc;
    }
    for (int idx = t; idx < 1536; idx += 1024) s_sh[idx] = s[b * 1536 + idx];
    __syncthreads();

    const float A  = -__expf(A_log[d * 16 + n]);
    const float dw = dt_w[d], db = dt_b[d];
    const float iw = in_w[d], ib = in_b[d];
    float h = 0.0f;
    float* yb = ybuf + (long)blockIdx.x * 1536 * 64;

    for (int step = 0; step < 1536; ++step) {
        const int c = dir ? (1535 - step) : step;
        const float sc  = s_sh[c];
        const float dtr = fmaf(sc, P1s[0], P0s[0]);
        const float xx  = fmaf(dtr, dw, db);
        const float dt  = (xx > 20.0f) ? xx : log1pf(__expf(xx));
        const float u   = fmaf(sc, iw, ib);
        const float Bv  = fmaf(sc, P1s[1 + n], P0s[1 + n]);
        const float Cv  = fmaf(sc, P1s[17 + n], P0s[17 + n]);
        h = __expf(dt * A) * h + dt * u * Bv;
        float contrib = h * Cv;
        contrib += __shfl_xor(contrib, 1);
        contrib += __shfl_xor(contrib, 2);
        contrib += __shfl_xor(contrib, 4);
        contrib += __shfl_xor(contrib, 8);
        if (n == 0) yb[(long)c * 64 + d] = contrib;
    }
}

__global__ void gate_kernel(const float* __restrict__ ybuf, const float* __restrict__ s,
                            const float* __restrict__ in_w, const float* __restrict__ in_b,
                            const float* __restrict__ Dp, const float* __restrict__ out_w,
                            const float* __restrict__ out_b, float* __restrict__ gate) {
    const int idx = blockIdx.x * 256 + threadIdx.x;
    if (idx >= 3072) return;
    const int b = idx / 1536, c = idx % 1536;
    const float sc = s[idx];
    const float* y0 = ybuf + ((long)(0 + b) * 1536 + c) * 64;
    const float* y1 = ybuf + ((long)(2 + b) * 1536 + c) * 64;
    float acc = out_b[0];
    for (int d = 0; d < 64; ++d) {
        const float u = fmaf(sc, in_w[d], in_b[d]);
        const float y = y0[d] + y1[d] + Dp[d] * u;
        acc = fmaf(y, out_w[d], acc);
    }
    gate[idx] = 1.0f / (1.0f + __expf(-acc));
}

// ---------------------------------------------------------------------------
// Fused: gated fuse GEMM (1536->384, fp32 WMMA) + BN + ReLU + heads (384->6, 384->2).
// Block = 256 thr (8 waves), 16 pixels x all 384 outputs.
// Double-buffered ASYNC global->LDS staging (6 chunks of 256 channels, 2 in
// flight, ASYNCcnt-retired) so HBM reads of cc overlap the WMMA stream.
// LDS pitch 260 (px-stride mod 64 banks = 4) -> conflict-free A reads.
// Gate is folded into the A fragment (per-channel) after ds_read.
// ---------------------------------------------------------------------------
__global__ void fuse_head_kernel(const float* __restrict__ cc, const float* __restrict__ gate,
                                 const float* __restrict__ fuse_w,
                                 const float* __restrict__ bn_g, const float* __restrict__ bn_b,
                                 const float* __restrict__ bn_m, const float* __restrict__ bn_v,
                                 const float* __restrict__ pred_w, const float* __restrict__ pred_b,
                                 const float* __restrict__ bird_w, const float* __restrict__ bird_b,
                                 float* __restrict__ out) {
    __shared__ float Xs0[16 * 260];
    __shared__ float Xs1[16 * 260];
    __shared__ float Hs[16 * 388];
    __shared__ float Gs[1536];

    const int tid  = threadIdx.x;
    const int wave = tid >> 5;
    const int lane = tid & 31;
    const int lo   = lane & 15;
    const int hi   = lane >> 4;

    const int b  = blockIdx.z;
    const int p0 = blockIdx.x * 16;

    // stage gate vector once (normal loads)
    for (int i = tid; i < 1536; i += 256) Gs[i] = gate[b * 1536 + i];

    // async-stage one 16px x 256ch chunk: 1024 x b128, 4 per thread
    auto stage = [&](int ch, float* Xbuf) {
#pragma unroll
        for (int k = 0; k < 4; ++k) {
            const int slot = tid + k * 256;
            const int px = slot >> 6;
            const int cs = (slot & 63) * 4;
            const float* g = cc + ((long)b * 16384 + p0 + px) * 1536 + ch * 256 + cs;
            float* l = Xbuf + px * 260 + cs;
#if USE_ASYNC_LDS
            __builtin_amdgcn_global_load_async_to_lds_b128(
                (__attribute__((address_space(1))) v4i*)(unsigned long long)g,
                (__attribute__((address_space(3))) v4i*)(unsigned long long)l, 0, 0);
#else
            *(float4*)l = *(const float4*)g;
#endif
        }
    };

    v8f acc[3] = {};

    stage(0, Xs0);
    stage(1, Xs1);

    for (int ch = 0; ch < 6; ++ch) {
        float* Xcur = (ch & 1) ? Xs1 : Xs0;
#if USE_ASYNC_LDS
        if (ch < 5) __builtin_amdgcn_s_wait_asynccnt(4);   // chunk ch landed, next still in flight
        else        __builtin_amdgcn_s_wait_asynccnt(0);
#endif
        __syncthreads();   // cross-wave visibility of chunk ch (and Gs on ch==0)

        const int cgbase = ch * 256;
        for (int c0 = 0; c0 < 256; c0 += 4) {
            v2f a;
            const float* ap = &Xcur[lo * 260 + c0 + 2 * hi];
            const float g0 = Gs[cgbase + c0 + 2 * hi];
            const float g1 = Gs[cgbase + c0 + 2 * hi + 1];
            a.x = ap[0] * g0;
            a.y = ap[1] * g1;
            const int cg0 = cgbase + c0;
#pragma unroll
            for (int t = 0; t < 3; ++t) {
                const int n = wave * 48 + t * 16 + lo;
                v2f bb;
                bb.x = fuse_w[(long)(cg0 + 2 * hi) * 384 + n];
                bb.y = fuse_w[(long)(cg0 + 2 * hi + 1) * 384 + n];
                acc[t] = __builtin_amdgcn_wmma_f32_16x16x4_f32(
                    false, a, false, bb, (short)0, acc[t], false, false);
            }
        }

        __syncthreads();   // all waves done reading Xcur before it is restaged
        if (ch + 2 < 6) stage(ch + 2, Xcur);
    }

    // BN + ReLU -> Hs
#pragma unroll
    for (int t = 0; t < 3; ++t) {
        const int n = wave * 48 + t * 16 + lo;
        const float scale = bn_g[n] * rsqrtf(bn_v[n] + 1e-5f);
        const float shift = bn_b[n] - bn_m[n] * scale;
#pragma unroll
        for (int r = 0; r < 8; ++r) {
            const int p = r + 8 * hi;
            float hv = fmaf(acc[t][r], scale, shift);
            Hs[p * 388 + n] = fmaxf(hv, 0.0f);
        }
    }
    __syncthreads();

    // heads: 16 pixels x 8 outputs (6 pred + 2 bird)
    if (tid < 128) {
        const int p = tid >> 3;
        const int j = tid & 7;
        float accv = (j < 6) ? pred_b[j] : bird_b[j - 6];
        for (int o = 0; o < 384; ++o) {
            const float hv = Hs[p * 388 + o];
            const float wv = (j < 6) ? pred_w[o * 6 + j] : bird_w[o * 2 + (j - 6)];
            accv = fmaf(hv, wv, accv);
        }
        const int pg = p0 + p;
        if (j < 6)
            out[((long)b * 6 + j) * 16384 + pg] = accv;
        else
            out[196608 + ((long)b * 2 + (j - 6)) * 16384 + pg] = accv;
    }
}

// ---------------------------------------------------------------------------
extern "C" void kernel_launch(void* const* d_in, const int* in_sizes, int n_in,
                              void* d_out, int out_size, void* d_ws, size_t ws_size,
                              hipStream_t stream) {
    const float* c1 = (const float*)d_in[0];
    const float* c2 = (const float*)d_in[1];
    const float* c3 = (const float*)d_in[2];
    const float* c4 = (const float*)d_in[3];
    const float* mlp1_w = (const float*)d_in[4];  const float* mlp1_b = (const float*)d_in[5];
    const float* mlp2_w = (const float*)d_in[6];  const float* mlp2_b = (const float*)d_in[7];
    const float* mlp3_w = (const float*)d_in[8];  const float* mlp3_b = (const float*)d_in[9];
    const float* mlp4_w = (const float*)d_in[10]; const float* mlp4_b = (const float*)d_in[11];
    const float* up2_w  = (const float*)d_in[12]; const float* up2_b  = (const float*)d_in[13];
    const float* up4_w  = (const float*)d_in[14]; const float* up4_b  = (const float*)d_in[15];
    const float* up8_w  = (const float*)d_in[16]; const float* up8_b  = (const float*)d_in[17];
    const float* ssm_in_w = (const float*)d_in[18];
    const float* ssm_in_b = (const float*)d_in[19];
    const float* ssm_xproj_w = (const float*)d_in[20];
    const float* ssm_dt_w = (const float*)d_in[21];
    const float* ssm_dt_b = (const float*)d_in[22];
    const float* ssm_A_log = (const float*)d_in[23];
    const float* ssm_D = (const float*)d_in[24];
    const float* ssm_out_w = (const float*)d_in[25];
    const float* ssm_out_b = (const float*)d_in[26];
    const float* fuse_w = (const float*)d_in[27];
    const float* bn_g = (const float*)d_in[28];
    const float* bn_b = (const float*)d_in[29];
    const float* bn_m = (const float*)d_in[30];
    const float* bn_v = (const float*)d_in[31];
    const float* pred_w = (const float*)d_in[32]; const float* pred_b = (const float*)d_in[33];
    const float* bird_w = (const float*)d_in[34]; const float* bird_b = (const float*)d_in[35];

    float* ws = (float*)d_ws;
    // workspace layout (floats)
    float* cc  = ws;                                // 2*16384*1536
    long off = 2L * 16384 * 1536;
    float* P2  = ws + off; off += 2L * 4096 * 384;  // proj(c2) pixel-major
    float* P3  = ws + off; off += 2L * 1024 * 384;
    float* P4  = ws + off; off += 2L * 256 * 384;
    float* O8  = ws + off; off += 2L * 256 * 512;   // offset convs
    float* O4  = ws + off; off += 2L * 1024 * 128;
    float* O2  = ws + off; off += 2L * 4096 * 32;
    float* part= ws + off; off += 16L * 2 * 1536;
    float* sbuf= ws + off; off += 2L * 1536;
    float* ybuf= ws + off; off += 4L * 1536 * 64;
    float* gate= ws + off; off += 2L * 1536;
    (void)ws_size; (void)n_in; (void)in_sizes; (void)out_size;

    // --- projections (WMMA, A = NCHW) ---
    wmma_gemm_kernel<true, 4><<<dim3(16384 / 128, 384 / 64, 2), 256, 0, stream>>>(
        c1, 96L * 16384, 16384, 0, mlp1_w, mlp1_b,
        cc + 1152, 16384L * 1536, 1536, 96, 384);
    wmma_gemm_kernel<true, 4><<<dim3(4096 / 128, 384 / 64, 2), 256, 0, stream>>>(
        c2, 96L * 4096, 4096, 0, mlp2_w, mlp2_b, P2, 4096L * 384, 384, 96, 384);
    wmma_gemm_kernel<true, 4><<<dim3(1024 / 128, 384 / 64, 2), 256, 0, stream>>>(
        c3, 192L * 1024, 1024, 0, mlp3_w, mlp3_b, P3, 1024L * 384, 384, 192, 384);
    wmma_gemm_kernel<true, 4><<<dim3(256 / 128, 384 / 64, 2), 256, 0, stream>>>(
        c4, 384L * 256, 256, 0, mlp4_w, mlp4_b, P4, 256L * 384, 384, 384, 384);

    // --- offset convs (WMMA, A = pixel-major proj buffers) ---
    wmma_gemm_kernel<false, 4><<<dim3(256 / 128, 512 / 64, 2), 256, 0, stream>>>(
        P4, 256L * 384, 256, 384, up8_w, up8_b, O8, 256L * 512, 512, 384, 512);
    wmma_gemm_kernel<false, 4><<<dim3(1024 / 128, 128 / 64, 2), 256, 0, stream>>>(
        P3, 1024L * 384, 1024, 384, up4_w, up4_b, O4, 1024L * 128, 128, 384, 128);
    wmma_gemm_kernel<false, 2><<<dim3(4096 / 128, 1, 2), 256, 0, stream>>>(
        P2, 4096L * 384, 4096, 384, up2_w, up2_b, O2, 4096L * 32, 32, 384, 32);

    // --- DySample into cc ---
    const long dtot = 2L * 128 * 128 * 4 * 96;
    const int dblocks = (int)(dtot / 256);
    dysample_kernel<<<dblocks, 256, 0, stream>>>(P4, O8, cc, 16, 16, 8, 0);
    dysample_kernel<<<dblocks, 256, 0, stream>>>(P3, O4, cc, 32, 32, 4, 384);
    dysample_kernel<<<dblocks, 256, 0, stream>>>(P2, O2, cc, 64, 64, 2, 768);

    // --- channel means ---
    colsum_partial_kernel<<<dim3(6, 16, 2), 256, 0, stream>>>(cc, part);
    colsum_reduce_kernel<<<12, 256, 0, stream>>>(part, sbuf);

    // --- SSM scan + gate ---
    ssm_scan_kernel<<<4, 1024, 0, stream>>>(sbuf, ssm_in_w, ssm_in_b, ssm_xproj_w,
                                            ssm_dt_w, ssm_dt_b, ssm_A_log, ybuf);
    gate_kernel<<<12, 256, 0, stream>>>(ybuf, sbuf, ssm_in_w, ssm_in_b, ssm_D,
                                        ssm_out_w, ssm_out_b, gate);

    // --- fused gated fuse-GEMM + BN + ReLU + heads ---
    fuse_head_kernel<<<dim3(1024, 1, 2), 256, 0, stream>>>(
        cc, gate, fuse_w, bn_g, bn_b, bn_m, bn_v,
        pred_w, pred_b, bird_w, bird_b, (float*)d_out);
}